// HierarchicalRLDialogueManager_26036091749066
// MI455X (gfx1250) — compile-verified
//
#include <hip/hip_runtime.h>

#define N_NODES 100000
#define N_EDGES 1000000
#define D_FEAT  128
#define N_ACT   64

typedef __attribute__((ext_vector_type(2))) float v2f;
typedef __attribute__((ext_vector_type(8))) float v8f;

// ---------------- degree / normalization ----------------

__global__ void k_init_deg(float* __restrict__ deg) {
    int i = blockIdx.x * blockDim.x + threadIdx.x;
    if (i < N_NODES) deg[i] = 1.0f;   // self-loop
}

__global__ void k_edge_deg(const int* __restrict__ ei, float* __restrict__ deg) {
    int e = blockIdx.x * blockDim.x + threadIdx.x;
    if (e < N_EDGES) {
        int dst = ei[N_EDGES + e];
        unsafeAtomicAdd(&deg[dst], 1.0f);
    }
}

__global__ void k_rsqrt(float* __restrict__ deg) {
    int i = blockIdx.x * blockDim.x + threadIdx.x;
    if (i < N_NODES) deg[i] = rsqrtf(deg[i]);   // deg -> dinv in place
}

// ---------------- h = x @ W_gcn  (fp32 WMMA) ----------------
// One block per 16-node row panel; 8 waves, one 16-col tile each.

__global__ __launch_bounds__(256) void k_gemm_h(const float* __restrict__ x,
                                                const float* __restrict__ W,
                                                float* __restrict__ h) {
    __shared__ float As[16 * 132];           // 16x128 panel, padded stride
    const int rowBase = blockIdx.x * 16;
    const int tid = threadIdx.x;

    // cooperative load: 16 rows x 32 float4 = 512 float4 loads, 2 per thread
    #pragma unroll
    for (int i = tid; i < 512; i += 256) {
        int r  = i >> 5;
        int c4 = i & 31;
        float4 v = ((const float4*)(x + (size_t)(rowBase + r) * D_FEAT))[c4];
        float* d = &As[r * 132 + c4 * 4];
        d[0] = v.x; d[1] = v.y; d[2] = v.z; d[3] = v.w;
    }
    __syncthreads();

    const int wave = tid >> 5;            // col tile 0..7
    const int lane = tid & 31;
    const int half = lane >> 4;           // 0: K pair {0,1}, 1: K pair {2,3}
    const int ln   = lane & 15;
    const int colBase = wave * 16;

    v8f acc = {};
    #pragma unroll
    for (int k = 0; k < D_FEAT; k += 4) {
        v2f a, b;
        // A 16x4 layout: lanes0-15 M=ln K={k,k+1}; lanes16-31 M=ln K={k+2,k+3}
        a.x = As[ln * 132 + k + 2 * half + 0];
        a.y = As[ln * 132 + k + 2 * half + 1];
        // B 4x16 layout: vgpr v, half h -> row K = k + 2*h + v, col = ln
        b.x = W[(size_t)(k + 2 * half + 0) * D_FEAT + colBase + ln];
        b.y = W[(size_t)(k + 2 * half + 1) * D_FEAT + colBase + ln];
        acc = __builtin_amdgcn_wmma_f32_16x16x4_f32(
            false, a, false, b, (short)0, acc, false, false);
    }

    // D 16x16 layout: vgpr v -> row m = v + 8*half, col = ln
    #pragma unroll
    for (int v = 0; v < 8; ++v)
        h[(size_t)(rowBase + v + 8 * half) * D_FEAT + colBase + ln] = acc[v];
}

// ---------------- agg init with fused self-loop ----------------

__global__ void k_selfloop(const float* __restrict__ h,
                           const float* __restrict__ dinv,
                           float* __restrict__ agg) {
    int t = blockIdx.x * blockDim.x + threadIdx.x;   // one float4 per thread
    if (t < N_NODES * 32) {
        int node = t >> 5;
        int c4   = t & 31;
        float s = dinv[node]; s = s * s;             // self-loop norm
        float4 v = ((const float4*)(h + (size_t)node * D_FEAT))[c4];
        v.x *= s; v.y *= s; v.z *= s; v.w *= s;
        ((float4*)(agg + (size_t)node * D_FEAT))[c4] = v;
    }
}

// ---------------- edge scatter: agg[dst] += h[src] * norm ----------------
// One wave per edge; lanes each own one float4 (32*4 = 128 feats).
// Edge endpoints forced into SGPRs so norm is scalar math and the
// gather/atomic bases use scalar addressing.

__global__ __launch_bounds__(256) void k_scatter(const int* __restrict__ ei,
                                                 const float* __restrict__ h,
                                                 const float* __restrict__ dinv,
                                                 float* __restrict__ agg) {
    const int e    = blockIdx.x * 8 + (threadIdx.x >> 5);  // exact: 8 waves/block
    const int lane = threadIdx.x & 31;
    const int src = __builtin_amdgcn_readfirstlane(ei[e]);
    const int dst = __builtin_amdgcn_readfirstlane(ei[N_EDGES + e]);
    const float norm = dinv[src] * dinv[dst];              // wave-uniform
    float4 v = ((const float4*)(h + (size_t)src * D_FEAT))[lane];
    float* base = agg + (size_t)dst * D_FEAT + (lane << 2);
    unsafeAtomicAdd(base + 0, v.x * norm);
    unsafeAtomicAdd(base + 1, v.y * norm);
    unsafeAtomicAdd(base + 2, v.z * norm);
    unsafeAtomicAdd(base + 3, v.w * norm);
}

// ---------------- heads: out = (agg + b_gcn) @ W_{high,low} + b ----------------
// Block per row panel; waves 0-3 -> high tiles, waves 4-7 -> low tiles.

__global__ __launch_bounds__(256) void k_heads(const float* __restrict__ agg,
                                               const float* __restrict__ bg,
                                               const float* __restrict__ Wh,
                                               const float* __restrict__ bh,
                                               const float* __restrict__ Wl,
                                               const float* __restrict__ bl,
                                               float* __restrict__ out) {
    __shared__ float As[16 * 132];
    const int rowBase = blockIdx.x * 16;
    const int tid = threadIdx.x;

    #pragma unroll
    for (int i = tid; i < 512; i += 256) {
        int r  = i >> 5;
        int c4 = i & 31;
        float4 v = ((const float4*)(agg + (size_t)(rowBase + r) * D_FEAT))[c4];
        float4 b = ((const float4*)bg)[c4];
        float* d = &As[r * 132 + c4 * 4];
        d[0] = v.x + b.x; d[1] = v.y + b.y; d[2] = v.z + b.z; d[3] = v.w + b.w;
    }
    __syncthreads();

    const int wave = tid >> 5;
    const int lane = tid & 31;
    const int half = lane >> 4;
    const int ln   = lane & 15;
    const int head = wave >> 2;                 // 0=high, 1=low
    const int colBase = (wave & 3) * 16;
    const float* W    = head ? Wl : Wh;
    const float* bias = head ? bl : bh;
    float* o = out + (size_t)head * N_NODES * N_ACT;

    v8f acc = {};
    #pragma unroll
    for (int k = 0; k < D_FEAT; k += 4) {
        v2f a, b;
        a.x = As[ln * 132 + k + 2 * half + 0];
        a.y = As[ln * 132 + k + 2 * half + 1];
        b.x = W[(size_t)(k + 2 * half + 0) * N_ACT + colBase + ln];
        b.y = W[(size_t)(k + 2 * half + 1) * N_ACT + colBase + ln];
        acc = __builtin_amdgcn_wmma_f32_16x16x4_f32(
            false, a, false, b, (short)0, acc, false, false);
    }

    float bv = bias[colBase + ln];
    #pragma unroll
    for (int v = 0; v < 8; ++v)
        o[(size_t)(rowBase + v + 8 * half) * N_ACT + colBase + ln] = acc[v] + bv;
}

// ---------------- launch ----------------

extern "C" void kernel_launch(void* const* d_in, const int* in_sizes, int n_in,
                              void* d_out, int out_size, void* d_ws, size_t ws_size,
                              hipStream_t stream) {
    const float* x     = (const float*)d_in[0];
    const int*   ei    = (const int*)d_in[1];
    const float* W_gcn = (const float*)d_in[2];
    const float* b_gcn = (const float*)d_in[3];
    const float* W_high= (const float*)d_in[4];
    const float* b_high= (const float*)d_in[5];
    const float* W_low = (const float*)d_in[6];
    const float* b_low = (const float*)d_in[7];
    float* out = (float*)d_out;

    // workspace layout: h | agg | dinv  (~103 MB)
    float* h    = (float*)d_ws;
    float* agg  = h   + (size_t)N_NODES * D_FEAT;
    float* dinv = agg + (size_t)N_NODES * D_FEAT;

    const int ROW_TILES = N_NODES / 16;   // 6250, exact

    k_init_deg<<<(N_NODES + 255) / 256, 256, 0, stream>>>(dinv);
    k_edge_deg<<<(N_EDGES + 255) / 256, 256, 0, stream>>>(ei, dinv);
    k_rsqrt  <<<(N_NODES + 255) / 256, 256, 0, stream>>>(dinv);

    k_gemm_h<<<ROW_TILES, 256, 0, stream>>>(x, W_gcn, h);

    k_selfloop<<<(N_NODES * 32 + 255) / 256, 256, 0, stream>>>(h, dinv, agg);
    k_scatter <<<N_EDGES / 8, 256, 0, stream>>>(ei, h, dinv, agg);

    k_heads<<<ROW_TILES, 256, 0, stream>>>(agg, b_gcn, W_high, b_high,
                                           W_low, b_low, out);
}